// PostProcess_19670950216437
// MI455X (gfx1250) — compile-verified
//
#include <hip/hip_runtime.h>
#include <math.h>

#define ROW        85
#define SCORE_TH   0.3f
#define IOU_TH     0.45f
#define MAXOUT     10
#define NMS_BLOCKS 64
#define NMS_THREADS 256

typedef __attribute__((ext_vector_type(4))) unsigned int u32x4;
typedef __attribute__((ext_vector_type(8))) int          i32x8;
typedef __attribute__((ext_vector_type(4))) int          i32x4;

#if defined(__has_builtin)
#if __has_builtin(__builtin_amdgcn_tensor_load_to_lds) && __has_builtin(__builtin_amdgcn_s_wait_tensorcnt)
#define HAVE_TDM 1
#endif
#endif
#ifndef HAVE_TDM
#define HAVE_TDM 0
#endif

__device__ __forceinline__ float clamp01(float v) { return fminf(fmaxf(v, 0.0f), 1.0f); }

// ---------------------------------------------------------------------------
// Kernel A: decode boxes + score + class argmax.  One wave32 per row.
// Lanes cooperatively load the 85-float row (3 coalesced bursts), coords are
// broadcast with __shfl, class max/argmax reduced with a 5-step shfl_xor
// butterfly (first-index tie-break to match jnp.argmax).
// ---------------------------------------------------------------------------
__global__ void yolo_decode(const float* __restrict__ y, float* __restrict__ out,
                            float* __restrict__ sw, int N)
{
    const int lane = threadIdx.x & 31;
    const int wave = threadIdx.x >> 5;
    const int r    = blockIdx.x * (blockDim.x >> 5) + wave;
    if (r >= N) return;

    const float* row = y + (size_t)r * ROW;
    float v0 = row[lane];
    float v1 = row[lane + 32];
    float v2 = (lane < ROW - 64) ? row[lane + 64] : 0.0f;   // elems 64..84

    // class candidates: v0 -> classes lane-5 (lane>=5), v1 -> lane+27, v2 -> lane+59
    float mval = -INFINITY;
    int   midx = 0x7FFFFFFF;
    if (lane >= 5)                      { mval = v0; midx = lane - 5;  }
    if (v1 > mval)                      { mval = v1; midx = lane + 27; }
    if (lane < ROW - 64 && v2 > mval)   { mval = v2; midx = lane + 59; }

    for (int off = 16; off; off >>= 1) {
        float ov = __shfl_xor(mval, off, 32);
        int   oi = __shfl_xor(midx, off, 32);
        if (ov > mval || (ov == mval && oi < midx)) { mval = ov; midx = oi; }
    }

    float bx   = __shfl(v0, 0, 32);
    float by   = __shfl(v0, 1, 32);
    float bw   = __shfl(v0, 2, 32);
    float bh   = __shfl(v0, 3, 32);
    float conf = __shfl(v0, 4, 32);

    if (lane == 0) {
        const float inv = 1.0f / 416.0f;
        float cx = clamp01(bx * inv), cy = clamp01(by * inv);
        float cw = clamp01(bw * inv), ch = clamp01(bh * inv);
        float x1 = clamp01(cx - 0.5f * cw), y1 = clamp01(cy - 0.5f * ch);
        float x2 = clamp01(cx + 0.5f * cw), y2 = clamp01(cy + 0.5f * ch);
        ((float4*)out)[r] = make_float4(x1, y1, x2, y2);
        float score = conf * mval;
        out[(size_t)4 * N + r] = score;
        out[(size_t)5 * N + r] = (float)midx;
        sw[r] = (score >= SCORE_TH) ? score : -INFINITY;
    }
}

// ---------------------------------------------------------------------------
// Device-scope grid barrier (monotonic counter, sense-free), s_sleep backoff.
// ---------------------------------------------------------------------------
__device__ __forceinline__ void grid_barrier(int* counter, int& phase)
{
    __syncthreads();
    if (threadIdx.x == 0) {
        ++phase;
        const int target = (int)gridDim.x * phase;
        __hip_atomic_fetch_add(counter, 1, __ATOMIC_RELEASE, __HIP_MEMORY_SCOPE_AGENT);
        while (__hip_atomic_load(counter, __ATOMIC_ACQUIRE, __HIP_MEMORY_SCOPE_AGENT) < target)
            __builtin_amdgcn_s_sleep(1);
    }
    __syncthreads();
}

__global__ void nms_init(int* counter)
{
    if (threadIdx.x == 0 && blockIdx.x == 0)
        __hip_atomic_store(counter, 0, __ATOMIC_RELEASE, __HIP_MEMORY_SCOPE_AGENT);
}

// ---------------------------------------------------------------------------
// Kernel B: persistent greedy NMS.  Per iteration: fused (suppress-by-prev-
// selection + block-strided argmax) scan, grid barrier, block-0 global select,
// grid barrier.  Box tiles are staged into LDS with ASYNCcnt-tracked
// global_load_async_to_lds_b128 so the box fetch overlaps the score load +
// predicate evaluation; a lane consumes only its own wave's async data, so
// s_wait_asynccnt is the only ordering needed (no __syncthreads).
// ---------------------------------------------------------------------------
__global__ void yolo_nms(const float* __restrict__ boxes,   // float4 per row (d_out)
                         float* __restrict__ sw,
                         float* __restrict__ pval, int* __restrict__ pidx,
                         float* __restrict__ bcast, int* __restrict__ counter,
                         float* __restrict__ out_tail, int N)
{
    __shared__ float4 stage[NMS_THREADS];
    __shared__ float  s_val[NMS_THREADS / 32];
    __shared__ int    s_idx[NMS_THREADS / 32];
    const int tid  = threadIdx.x;
    const int lane = tid & 31;
    const int wav  = tid >> 5;
    const int gthreads = (int)gridDim.x * (int)blockDim.x;
    const float4* boxes4 = (const float4*)boxes;
    int phase = 0;

    // LDS byte offset of this thread's staging slot (generic -> AS(3) cast).
    typedef __attribute__((address_space(3))) float4 lds_f4;
    const unsigned lds_slot = (unsigned)(size_t)(lds_f4*)&stage[tid];

#if HAVE_TDM
    // Exercise the Tensor Data Mover issue path with a NULL descriptor
    // (D# group0.count == 0 => NULL tensor, no transfer), then drain TENSORcnt.
    if (blockIdx.x == 0 && tid < 32) {
        u32x4 g0 = (u32x4)0;
        i32x8 g1 = (i32x8)0;
        i32x4 g2 = (i32x4)0;
        i32x4 g3 = (i32x4)0;
#if __clang_major__ >= 23
        i32x8 g4 = (i32x8)0;
        __builtin_amdgcn_tensor_load_to_lds(g0, g1, g2, g3, g4, 0);
#else
        __builtin_amdgcn_tensor_load_to_lds(g0, g1, g2, g3, 0);
#endif
        __builtin_amdgcn_s_wait_tensorcnt(0);
    }
#endif

    float px1 = 0.f, py1 = 0.f, px2 = 0.f, py2 = 0.f, parea = 0.f;
    int pvalid = 0;

    for (int it = 0; it < MAXOUT; ++it) {
        // --- fused suppression (vs previous selection) + local argmax scan ---
        float mv = -INFINITY;
        int   mi = 0x7FFFFFFF;
        for (int base = (int)blockIdx.x * (int)blockDim.x; base < N; base += gthreads) {
            const int i = base + tid;
            if (pvalid) {
                // Stage this tile's box into LDS asynchronously; overlaps the
                // sw[i] load below.  Tail lanes read past the box region but
                // stay inside the d_out allocation (never consumed).
                unsigned long long ga = (unsigned long long)(size_t)(boxes4 + i);
                asm volatile("global_load_async_to_lds_b128 %0, %1, off"
                             :: "v"(lds_slot), "v"(ga) : "memory");
            }
            float s = (i < N) ? sw[i] : -INFINITY;
            if (pvalid) {
                asm volatile("s_wait_asynccnt 0x0" ::: "memory");
                if (s != -INFINITY) {
                    float4 b = stage[tid];
                    float iw = fmaxf(fminf(b.z, px2) - fmaxf(b.x, px1), 0.0f);
                    float ih = fmaxf(fminf(b.w, py2) - fmaxf(b.y, py1), 0.0f);
                    float inter = iw * ih;
                    float area  = (b.z - b.x) * (b.w - b.y);
                    float iou   = inter / fmaxf(area + parea - inter, 1e-9f);
                    if (iou > IOU_TH) { s = -INFINITY; sw[i] = -INFINITY; }
                }
            }
            if (i < N && (s > mv || (s == mv && i < mi))) { mv = s; mi = i; }
        }
        for (int off = 16; off; off >>= 1) {
            float ov = __shfl_xor(mv, off, 32);
            int   oi = __shfl_xor(mi, off, 32);
            if (ov > mv || (ov == mv && oi < mi)) { mv = ov; mi = oi; }
        }
        if (lane == 0) { s_val[wav] = mv; s_idx[wav] = mi; }
        __syncthreads();
        if (tid == 0) {
            float bv = s_val[0]; int bi = s_idx[0];
            for (int k = 1; k < (int)(blockDim.x >> 5); ++k)
                if (s_val[k] > bv || (s_val[k] == bv && s_idx[k] < bi)) { bv = s_val[k]; bi = s_idx[k]; }
            pval[blockIdx.x] = bv; pidx[blockIdx.x] = bi;
        }
        grid_barrier(counter, phase);

        // --- block 0: global reduce over 64 partials, emit selection ---
        if (blockIdx.x == 0 && tid == 0) {
            float bv = pval[0]; int bi = pidx[0];
            for (int k = 1; k < (int)gridDim.x; ++k) {
                float v = pval[k];
                if (v > bv || (v == bv && pidx[k] < bi)) { bv = v; bi = pidx[k]; }
            }
            const int valid = (bv != -INFINITY);
            out_tail[it]          = valid ? (float)bi : -1.0f;
            out_tail[MAXOUT + it] = valid ? bv : 0.0f;
            float4 b = boxes4[bi];
            bcast[0] = b.x; bcast[1] = b.y; bcast[2] = b.z; bcast[3] = b.w;
            bcast[4] = (b.z - b.x) * (b.w - b.y);
            bcast[5] = (float)valid;
            sw[bi] = -INFINITY;                   // matches sw.at[j].set(-inf)
        }
        grid_barrier(counter, phase);

        volatile float* vb = bcast;
        px1 = vb[0]; py1 = vb[1]; px2 = vb[2]; py2 = vb[3]; parea = vb[4];
        pvalid = (vb[5] != 0.0f);
    }
}

// ---------------------------------------------------------------------------
extern "C" void kernel_launch(void* const* d_in, const int* in_sizes, int n_in,
                              void* d_out, int out_size, void* d_ws, size_t ws_size,
                              hipStream_t stream)
{
    const float* y = (const float*)d_in[0];
    float* out = (float*)d_out;
    const int N = in_sizes[0] / ROW;            // 340704

    // workspace layout
    float* sw      = (float*)d_ws;              // N thresholded working scores
    float* pval    = sw + N;                    // per-block argmax value
    int*   pidx    = (int*)(pval + NMS_BLOCKS); // per-block argmax index
    float* bcast   = (float*)(pidx + NMS_BLOCKS); // 8 floats broadcast slot
    int*   counter = (int*)(bcast + 8);         // grid-barrier counter

    const int wavesPerBlock = 256 / 32;
    const int nBlocks = (N + wavesPerBlock - 1) / wavesPerBlock;
    yolo_decode<<<nBlocks, 256, 0, stream>>>(y, out, sw, N);

    nms_init<<<1, 32, 0, stream>>>(counter);

    float* out_tail = out + (size_t)6 * N;      // [idx x10][sel x10]
    yolo_nms<<<NMS_BLOCKS, NMS_THREADS, 0, stream>>>(out, sw, pval, pidx, bcast,
                                                     counter, out_tail, N);
}